// RegressionLoss_2310692405454
// MI455X (gfx1250) — compile-verified
//
#include <hip/hip_runtime.h>

typedef __attribute__((ext_vector_type(2))) float v2f;
typedef __attribute__((ext_vector_type(4))) float v4f;
typedef __attribute__((ext_vector_type(8))) float v8f;

#define CLASS_K 1024.0f   // class-mismatch penalty: >= 1024 >> 25+max dist^2; exact in f32 for c in {0..3}

// ---------------------------------------------------------------- zero scratch
__global__ void rl_zero_kernel(unsigned int* __restrict__ w, int nwords) {
    for (int i = blockIdx.x * blockDim.x + threadIdx.x; i < nwords;
         i += gridDim.x * blockDim.x)
        w[i] = 0u;
}

// ----------------------------------------------- precompute augmented A rows
// pred row p -> [-2x, -2y, x^2+y^2+K*c^2, -2K*c]  (16B, one b128 store/thread)
__global__ void rl_prep_kernel(const float* __restrict__ pred,
                               float* __restrict__ aug, int n) {
    int i = blockIdx.x * blockDim.x + threadIdx.x;
    if (i >= n) return;
    const float c = pred[3 * (size_t)i + 0];
    const float x = pred[3 * (size_t)i + 1];
    const float y = pred[3 * (size_t)i + 2];
    v4f r;
    r.x = -2.0f * x;
    r.y = -2.0f * y;
    r.z = x * x + y * y + CLASS_K * c * c;
    r.w = -2.0f * CLASS_K * c;
    *(v4f*)(aug + 4 * (size_t)i) = r;
}

// ------------------------------------------------- pairwise match via WMMA f32
// One wave per (batch b, tile of 16 targets). Each V_WMMA_F32_16X16X4_F32
// produces a 16x16 tile of  dist^2 + K*(dc)^2 ; the per-target constant
// |t|^2 + K*ct^2 rides in the C accumulator. Inner loop: b64 load + WMMA +
// running argmin (strict '<' preserves lowest-index ties like jnp.argmax).
__global__ __launch_bounds__(32) void rl_match_kernel(
    const float* __restrict__ aug, const float* __restrict__ gt,
    unsigned char* __restrict__ hits, int P, int T) {
    const int  lane = threadIdx.x;          // 0..31 (wave32)
    const int  b    = blockIdx.y;
    const int  n    = lane & 15;            // column (target) / row-in-tile index
    const bool hi   = lane >= 16;           // lanes 16..31: K=2,3 half / rows 8..15

    // Target for column n of this tile (gt layout: [class, x, y])
    const float* tg = gt + ((size_t)b * T + (blockIdx.x * 16 + n)) * 3;
    const float tc = tg[0], tx = tg[1], ty = tg[2];

    // B matrix (4x16 f32): lanes 0-15 -> K=0,1 = (x_t, y_t); lanes 16-31 -> K=2,3 = (1, c_t)
    v2f bmat;
    bmat.x = hi ? 1.0f : tx;
    bmat.y = hi ? tc   : ty;

    // C accumulator: per-column constant broadcast down all 8 rows
    const float tconst = tx * tx + ty * ty + CLASS_K * tc * tc;
    v8f cmat;
#pragma unroll
    for (int g = 0; g < 8; ++g) cmat[g] = tconst;

    float bestVal = 3.4e38f;
    int   bestIdx = 0;

    const int   hiofs = hi ? 8 : 0;
    // Per-lane A fetch: row (p0+n), K-pair (hi ? [2,3] : [0,1]) => single b64
    const float* abase = aug + ((size_t)b * P + n) * 4 + (hi ? 2 : 0);

    for (int p0 = 0; p0 < P; p0 += 16) {
        const v2f amat = *(const v2f*)(abase + (size_t)p0 * 4);

        const v8f d = __builtin_amdgcn_wmma_f32_16x16x4_f32(
            false, amat, false, bmat, (short)0, cmat, false, false);

        // D layout: lane<16 holds rows M=g; lane>=16 holds rows M=g+8.
        const int tbase = p0 + hiofs;
#pragma unroll
        for (int g = 0; g < 8; ++g)
            if (d[g] < bestVal) { bestVal = d[g]; bestIdx = tbase + g; }
    }

    // Single cross-lane combine (rows 0-7+16k vs 8-15+16k). Lowest index wins ties.
    const float ov = __shfl_xor(bestVal, 16, 32);
    const int   oi = __shfl_xor(bestIdx, 16, 32);
    if (ov < bestVal || (ov == bestVal && oi < bestIdx)) { bestVal = ov; bestIdx = oi; }

    // Valid match iff min penalized dist^2 <= RADIUS^2 (class mismatch >= 1024).
    // Racing stores all write 1 -> benign.
    if (!hi && bestVal <= 25.0f)
        hits[(size_t)b * P + bestIdx] = 1;
}

// --------------------------------------------------------------- count hits
__global__ void rl_count_kernel(const unsigned int* __restrict__ hitsW,
                                unsigned int* __restrict__ counter, int nwords) {
    unsigned int local = 0;
    for (int i = blockIdx.x * blockDim.x + threadIdx.x; i < nwords;
         i += gridDim.x * blockDim.x) {
        unsigned int x = hitsW[i];  // four 0/1 bytes
        local += (x & 0xFFu) + ((x >> 8) & 0xFFu) + ((x >> 16) & 0xFFu) + ((x >> 24) & 0xFFu);
    }
#pragma unroll
    for (int off = 16; off > 0; off >>= 1)
        local += __shfl_xor(local, off, 32);
    if ((threadIdx.x & 31) == 0 && local)
        atomicAdd(counter, local);
}

// --------------------------------------------------------------- final loss
__global__ void rl_finalize_kernel(const unsigned int* __restrict__ counter,
                                   float* __restrict__ out, float BP, float BT) {
    const float tp  = (float)(*counter);
    const float fp  = BP - tp;
    const float fn  = BT - tp;
    const float eps = 1e-6f;
    const float precision = (tp + eps) / (tp + eps + fp + eps);
    const float recall    = (tp + eps) / (tp + fn + eps);
    const float f1 = 2.0f * precision * recall / (precision + recall);
    out[0] = 1.0f - f1;
}

extern "C" void kernel_launch(void* const* d_in, const int* in_sizes, int n_in,
                              void* d_out, int out_size, void* d_ws, size_t ws_size,
                              hipStream_t stream) {
    const float* pred = (const float*)d_in[0];   // (B,P,3) f32
    const float* gt   = (const float*)d_in[1];   // (B,T,3) f32

    const int B  = 16;
    const int P  = in_sizes[0] / (B * 3);        // 4096
    const int T  = in_sizes[1] / (B * 3);        // 1024
    const int BP = B * P;

    // ws layout: [aug: BP*4 f32 = 1MB][hits: BP bytes][counter: u32]
    float*         aug     = (float*)d_ws;
    unsigned char* hits    = (unsigned char*)d_ws + (size_t)BP * 4 * sizeof(float);
    unsigned int*  counter = (unsigned int*)(hits + BP);

    const int hitWords = BP / 4;                 // BP is a multiple of 4
    rl_zero_kernel<<<64, 256, 0, stream>>>((unsigned int*)hits, hitWords + 1);
    rl_prep_kernel<<<(BP + 255) / 256, 256, 0, stream>>>(pred, aug, BP);

    dim3 grid(T / 16, B);                        // one wave per 16-target tile
    rl_match_kernel<<<grid, 32, 0, stream>>>(aug, gt, hits, P, T);

    rl_count_kernel<<<64, 256, 0, stream>>>((const unsigned int*)hits, counter, hitWords);

    rl_finalize_kernel<<<1, 1, 0, stream>>>(counter, (float*)d_out,
                                            (float)BP, (float)(B * T));
}